// GATLayerONNX_51994874085825
// MI455X (gfx1250) — compile-verified
//
#include <hip/hip_runtime.h>
#include <hip/hip_bf16.h>

typedef __attribute__((ext_vector_type(2))) float v2f;
typedef __attribute__((ext_vector_type(8))) float v8f;

#define IN_C 128
#define OUT_C 32
#define NEG_SLOPE 0.2f

// ---------------------------------------------------------------------------
// Kernel 1: Wx = x @ W via V_WMMA_F32_16X16X4_F32.
// One wave per 16-row M tile; two 16x16 accumulators cover N=32.
// W is staged in LDS once per block, PRE-SWIZZLED so that each lane's B
// operand (two consecutive K rows at one N column) is a contiguous float2:
//   Bs[(c*2 + kh)*32 + n] = { W[(4c+2kh)*32 + n], W[(4c+2kh+1)*32 + n] }
// -> the inner loop's B fetch is a single ds_load_b64 per operand, no repack.
// ---------------------------------------------------------------------------
__global__ __launch_bounds__(256) void gat_gemm_wmma(
    const float* __restrict__ x, const float* __restrict__ W,
    float* __restrict__ Wx, int n_nodes) {
  __shared__ v2f Bs[(IN_C / 4) * 2 * OUT_C];  // 2048 pairs = 16 KB
  const int t = threadIdx.x;
  for (int p = t; p < 2048; p += 256) {
    const int c = p >> 6;       // k-chunk (K = 4c)
    const int r = p & 63;
    const int kh = r >> 5;      // lane-half
    const int n = r & 31;       // output column
    const int k = 4 * c + 2 * kh;
    v2f bp;
    bp.x = W[k * OUT_C + n];
    bp.y = W[(k + 1) * OUT_C + n];
    Bs[p] = bp;
  }
  __syncthreads();

  const int wave = t >> 5;
  const int lane = t & 31;
  const int mTile = blockIdx.x * 8 + wave;
  if (mTile * 16 >= n_nodes) return;

  const int mBase = mTile * 16;
  const int nlane = lane & 15;   // row (A) / column (B,C) index within half-wave
  const int kh = lane >> 4;      // which half of the wave

  v8f acc0 = {};  // N = 0..15
  v8f acc1 = {};  // N = 16..31

  // A row as float2 pairs: pair index for k-chunk c is 2c + kh.
  const v2f* Arow = (const v2f*)(x + (size_t)(mBase + nlane) * IN_C);

  for (int c = 0; c < IN_C / 4; ++c) {
    const v2f a = Arow[2 * c + kh];
    const int base = (c * 2 + kh) << 5;
    const v2f b0 = Bs[base + nlane];
    const v2f b1 = Bs[base + 16 + nlane];

    acc0 = __builtin_amdgcn_wmma_f32_16x16x4_f32(false, a, false, b0,
                                                 (short)0, acc0, false, false);
    acc1 = __builtin_amdgcn_wmma_f32_16x16x4_f32(false, a, false, b1,
                                                 (short)0, acc1, false, false);
  }

  // C layout: VGPR j -> M = j + 8*kh, N = nlane (+16 for acc1).
  float* Crow = Wx + (size_t)mBase * OUT_C;
  #pragma unroll
  for (int j = 0; j < 8; ++j) {
    const int m = j + 8 * kh;
    Crow[m * OUT_C + nlane] = acc0[j];
    Crow[m * OUT_C + 16 + nlane] = acc1[j];
  }
}

// ---------------------------------------------------------------------------
// Kernel 2: per-node alpha_src/alpha_dst dot products + init of max/sum/out.
// One wave32 per node, lane = channel, shuffle reduction.
// ---------------------------------------------------------------------------
__global__ void gat_node_alpha_init(
    const float* __restrict__ Wx,
    const float* __restrict__ att_src, const float* __restrict__ att_dst,
    float* __restrict__ as_, float* __restrict__ ad_,
    float* __restrict__ mx, float* __restrict__ sm,
    float* __restrict__ out, int n_nodes) {
  const int gid = blockIdx.x * blockDim.x + threadIdx.x;
  const int node = __builtin_amdgcn_readfirstlane(gid >> 5);  // wave-uniform
  const int lane = gid & 31;
  if (node >= n_nodes) return;

  const float w = Wx[(size_t)node * OUT_C + lane];
  float vs = w * att_src[lane];
  float vd = w * att_dst[lane];
  #pragma unroll
  for (int off = 16; off > 0; off >>= 1) {
    vs += __shfl_xor(vs, off, 32);
    vd += __shfl_xor(vd, off, 32);
  }
  out[(size_t)node * OUT_C + lane] = 0.0f;
  if (lane == 0) {
    as_[node] = vs;
    ad_[node] = vd;
    mx[node] = -INFINITY;
    sm[node] = 0.0f;
  }
}

// Float atomic max via sign-split integer punning (valid with -inf init).
__device__ __forceinline__ void atomicMaxF32(float* addr, float v) {
  if (v >= 0.0f)
    atomicMax((int*)addr, __float_as_int(v));
  else
    atomicMin((unsigned int*)addr, (unsigned int)__float_as_int(v));
}

// ---------------------------------------------------------------------------
// Kernel 3: edge logits e = leakyrelu(ad[dst] + as[src]); segment max.
// ---------------------------------------------------------------------------
__global__ void gat_edge_logit_max(
    const int* __restrict__ src, const int* __restrict__ dst,
    const float* __restrict__ as_, const float* __restrict__ ad_,
    float* __restrict__ ev, float* __restrict__ mx, int n_edges) {
  const int i = blockIdx.x * blockDim.x + threadIdx.x;
  if (i >= n_edges) return;
  const int s = src[i], d = dst[i];
  float v = ad_[d] + as_[s];
  v = (v >= 0.0f) ? v : NEG_SLOPE * v;
  ev[i] = v;
  atomicMaxF32(&mx[d], v);
}

// ---------------------------------------------------------------------------
// Kernel 4: exp(e - max[dst]); segment sum. Overwrites ev with exp values.
// ---------------------------------------------------------------------------
__global__ void gat_edge_exp_sum(
    const int* __restrict__ dst, const float* __restrict__ mx,
    float* __restrict__ ev, float* __restrict__ sm, int n_edges) {
  const int i = blockIdx.x * blockDim.x + threadIdx.x;
  if (i >= n_edges) return;
  const int d = dst[i];
  const float ex = __expf(ev[i] - mx[d]);
  ev[i] = ex;
  atomicAdd(&sm[d], ex);
}

// ---------------------------------------------------------------------------
// Kernel 5: weighted message scatter. One wave32 per edge, lane = channel.
// Edge metadata is wave-uniform -> readfirstlane pushes it to the scalar
// path (s_load) instead of 32 redundant vector loads.
// out[dst,c] += (exp_e/sum[dst]) * Wx[src,c]  (f32 global atomics, L2-resident)
// ---------------------------------------------------------------------------
__global__ void gat_edge_scatter(
    const int* __restrict__ src, const int* __restrict__ dst,
    const float* __restrict__ ev, const float* __restrict__ sm,
    const float* __restrict__ Wx, float* __restrict__ out, int n_edges) {
  const int gid = blockIdx.x * blockDim.x + threadIdx.x;
  const int i = __builtin_amdgcn_readfirstlane(gid >> 5);  // wave-uniform edge id
  const int lane = gid & 31;
  if (i >= n_edges) return;
  const int s = src[i], d = dst[i];
  const float alpha = ev[i] / (sm[d] + 1e-16f);
  atomicAdd(&out[(size_t)d * OUT_C + lane], alpha * Wx[(size_t)s * OUT_C + lane]);
}

// ---------------------------------------------------------------------------
// Kernel 6: ELU, in place on output.
// ---------------------------------------------------------------------------
__global__ void gat_elu(float* __restrict__ out, int n) {
  const int i = blockIdx.x * blockDim.x + threadIdx.x;
  if (i >= n) return;
  const float h = out[i];
  out[i] = (h > 0.0f) ? h : expm1f(h);
}

// ---------------------------------------------------------------------------
extern "C" void kernel_launch(void* const* d_in, const int* in_sizes, int n_in,
                              void* d_out, int out_size, void* d_ws, size_t ws_size,
                              hipStream_t stream) {
  const float* x = (const float*)d_in[0];
  const int* edge_index = (const int*)d_in[1];
  const float* W = (const float*)d_in[2];
  const float* att_src = (const float*)d_in[3];
  const float* att_dst = (const float*)d_in[4];

  const int N = in_sizes[0] / IN_C;       // 100000
  const int E = in_sizes[1] / 2;          // 1600000
  const int* src = edge_index;
  const int* dst = edge_index + E;

  float* out = (float*)d_out;

  // Workspace layout (floats): Wx | as | ad | mx | sm | ev  (~20.8 MB)
  float* ws = (float*)d_ws;
  float* Wx = ws;
  float* as_ = Wx + (size_t)N * OUT_C;
  float* ad_ = as_ + N;
  float* mx = ad_ + N;
  float* sm = mx + N;
  float* ev = sm + N;

  // 1) GEMM via WMMA: 8 M-tiles (of 16 rows) per 256-thread block
  {
    const int mTiles = (N + 15) / 16;
    const int grid = (mTiles + 7) / 8;
    gat_gemm_wmma<<<grid, 256, 0, stream>>>(x, W, Wx, N);
  }
  // 2) node alphas + init
  {
    const int threads = N * 32;
    gat_node_alpha_init<<<(threads + 255) / 256, 256, 0, stream>>>(
        Wx, att_src, att_dst, as_, ad_, mx, sm, out, N);
  }
  // 3) edge logits + segment max
  gat_edge_logit_max<<<(E + 255) / 256, 256, 0, stream>>>(src, dst, as_, ad_, ev, mx, E);
  // 4) exp + segment sum
  gat_edge_exp_sum<<<(E + 255) / 256, 256, 0, stream>>>(dst, mx, ev, sm, E);
  // 5) scatter weighted messages (wave per edge)
  {
    const long long threads = (long long)E * 32;
    gat_edge_scatter<<<(int)((threads + 255) / 256), 256, 0, stream>>>(
        src, dst, ev, sm, Wx, out, E);
  }
  // 6) ELU
  gat_elu<<<(N * OUT_C + 255) / 256, 256, 0, stream>>>(out, N * OUT_C);
}